// WeightedGCNLayer_28346784154213
// MI455X (gfx1250) — compile-verified
//
#include <hip/hip_runtime.h>

typedef float v2f __attribute__((ext_vector_type(2)));
typedef float v8f __attribute__((ext_vector_type(8)));

#define D_DIM 128
#define BN_EPS 1e-5f

// ---------------------------------------------------------------------------
// Kernel 0: zero the accumulation region of d_out and the stats scratch.
// ---------------------------------------------------------------------------
__global__ __launch_bounds__(256) void zero_kernel(float* __restrict__ out,
                                                   float* __restrict__ stats,
                                                   long long n_out, int n_stats) {
    long long stride = (long long)gridDim.x * blockDim.x;
    for (long long i = (long long)blockIdx.x * blockDim.x + threadIdx.x;
         i < n_out; i += stride)
        out[i] = 0.0f;
    int gid = blockIdx.x * blockDim.x + threadIdx.x;
    if (gid < n_stats) stats[gid] = 0.0f;
}

// ---------------------------------------------------------------------------
// Kernel 1: XW = x @ W via V_WMMA_F32_16X16X4_F32.
// One wave per 16x16 output tile; 8 waves (256 threads) per block cover a
// 128-row slab; blockIdx.y picks the 16-wide column tile (D=128 -> 8 tiles).
// Fragment layouts per CDNA5 ISA 7.12.2 (32-bit A 16x4 / B 4x16 / C 16x16).
// ---------------------------------------------------------------------------
__global__ __launch_bounds__(256) void gemm_xw_wmma(const float* __restrict__ x,
                                                    const float* __restrict__ w,
                                                    float* __restrict__ xw, int N) {
    const int lane = threadIdx.x & 31;
    const int wave = threadIdx.x >> 5;
    const int m0 = (blockIdx.x * 8 + wave) * 16;
    if (m0 >= N) return;                     // wave-uniform: EXEC stays all-1s
    const int n0   = blockIdx.y * 16;
    const int hi   = lane >> 4;              // 0: lanes 0-15, 1: lanes 16-31
    const int mrow = m0 + (lane & 15);
    const int ncol = n0 + (lane & 15);

    v8f acc = {0.f, 0.f, 0.f, 0.f, 0.f, 0.f, 0.f, 0.f};
    #pragma unroll 4
    for (int k0 = 0; k0 < D_DIM; k0 += 4) {
        const int ka = k0 + (hi << 1);       // K pair handled by this half-wave
        v2f a;                               // A 16x4: row striped in lane
        a.x = x[(long long)mrow * D_DIM + ka];
        a.y = x[(long long)mrow * D_DIM + ka + 1];
        v2f b;                               // B 4x16: K-row striped across lanes
        b.x = w[ka * D_DIM + ncol];
        b.y = w[(ka + 1) * D_DIM + ncol];
        acc = __builtin_amdgcn_wmma_f32_16x16x4_f32(
            /*neg_a=*/false, a, /*neg_b=*/false, b,
            /*c_mod=*/(short)0, acc, /*reuse_a=*/false, /*reuse_b=*/false);
    }
    const int rbase = m0 + hi * 8;           // D: VGPR v -> rows v / v+8
    #pragma unroll
    for (int v = 0; v < 8; ++v)
        xw[(long long)(rbase + v) * D_DIM + ncol] = acc[v];
}

// ---------------------------------------------------------------------------
// Kernel 2: symmetric weighted scatter-add over edges (incl. self edges).
// One thread per (edge, feature) pair; f32 global atomics resolve in L2.
// ---------------------------------------------------------------------------
__global__ __launch_bounds__(256) void edge_scatter(const float* __restrict__ xw,
                                                    const int* __restrict__ edge_index,
                                                    const int* __restrict__ edge_type,
                                                    const float* __restrict__ alpha,
                                                    float* __restrict__ out,
                                                    int E, int N, int self_rel) {
    const long long total  = (long long)(E + N) * D_DIM;
    const long long stride = (long long)gridDim.x * blockDim.x;
    for (long long i = (long long)blockIdx.x * blockDim.x + threadIdx.x;
         i < total; i += stride) {
        const int e = (int)(i >> 7);
        const int t = (int)(i & (D_DIM - 1));
        int row, col, et;
        if (e < E) {
            row = edge_index[e];             // edge_index[0][e]
            col = edge_index[E + e];         // edge_index[1][e]
            et  = edge_type[e];
        } else {
            row = col = e - E;               // self edge
            et  = self_rel;
        }
        const float a  = alpha[et];
        const float xc = xw[(long long)col * D_DIM + t];
        const float xr = xw[(long long)row * D_DIM + t];
        atomicAdd(&out[(long long)row * D_DIM + t], a * xc);
        atomicAdd(&out[(long long)col * D_DIM + t], a * xr);
    }
}

// ---------------------------------------------------------------------------
// Kernel 3: per-column sum and sum-of-squares (for BatchNorm batch stats).
// blockDim = 256 (multiple of 128) -> each thread touches one fixed column.
// ---------------------------------------------------------------------------
__global__ __launch_bounds__(256) void col_stats(const float* __restrict__ out,
                                                 float* __restrict__ sums,
                                                 float* __restrict__ sumsq, int N) {
    __shared__ float ls[256];
    __shared__ float ls2[256];
    float s = 0.f, s2 = 0.f;
    const long long total  = (long long)N * D_DIM;
    const long long stride = (long long)gridDim.x * blockDim.x;  // mult. of 128
    for (long long i = (long long)blockIdx.x * blockDim.x + threadIdx.x;
         i < total; i += stride) {
        const float v = out[i];
        s  += v;
        s2 += v * v;
    }
    ls[threadIdx.x]  = s;
    ls2[threadIdx.x] = s2;
    __syncthreads();
    if (threadIdx.x < D_DIM) {
        atomicAdd(&sums[threadIdx.x],  ls[threadIdx.x]  + ls[threadIdx.x + D_DIM]);
        atomicAdd(&sumsq[threadIdx.x], ls2[threadIdx.x] + ls2[threadIdx.x + D_DIM]);
    }
}

// ---------------------------------------------------------------------------
// Kernel 4: BatchNorm normalize in place. (bias cancels inside (out - mean))
// ---------------------------------------------------------------------------
__global__ __launch_bounds__(256) void bn_normalize(float* __restrict__ out,
                                                    const float* __restrict__ sums,
                                                    const float* __restrict__ sumsq,
                                                    const float* __restrict__ gamma,
                                                    const float* __restrict__ beta,
                                                    int N) {
    const long long total  = (long long)N * D_DIM;
    const long long stride = (long long)gridDim.x * blockDim.x;
    const float invN = 1.0f / (float)N;
    for (long long i = (long long)blockIdx.x * blockDim.x + threadIdx.x;
         i < total; i += stride) {
        const int c = (int)(i & (D_DIM - 1));
        const float mean = sums[c] * invN;
        const float var  = sumsq[c] * invN - mean * mean;
        out[i] = (out[i] - mean) * rsqrtf(var + BN_EPS) * gamma[c] + beta[c];
    }
}

// ---------------------------------------------------------------------------
extern "C" void kernel_launch(void* const* d_in, const int* in_sizes, int n_in,
                              void* d_out, int out_size, void* d_ws, size_t ws_size,
                              hipStream_t stream) {
    (void)n_in; (void)out_size; (void)ws_size;

    const float* x          = (const float*)d_in[0];
    const float* r          = (const float*)d_in[1];
    const int*   edge_index = (const int*)d_in[2];
    const int*   edge_type  = (const int*)d_in[3];
    const float* weight     = (const float*)d_in[4];
    const float* alpha      = (const float*)d_in[5];
    const float* bias       = (const float*)d_in[6];
    const float* gamma      = (const float*)d_in[7];
    const float* beta       = (const float*)d_in[8];
    (void)bias;  // bias cancels algebraically inside BatchNorm's (out - mean)

    const int N        = in_sizes[0] / D_DIM;
    const int E        = in_sizes[2] / 2;
    const int NUM_REL  = in_sizes[1] / D_DIM;
    const int SELF_REL = NUM_REL - 1;

    float* out   = (float*)d_out;                 // [N, 128] normalized output
    float* xw    = (float*)d_ws;                  // [N, 128] scratch XW
    float* sums  = xw + (size_t)N * D_DIM;        // [128]
    float* sumsq = sums + D_DIM;                  // [128]

    const long long nOut = (long long)N * D_DIM;

    // 0) zero accumulator + stats
    zero_kernel<<<2048, 256, 0, stream>>>(out, sums, nOut, 2 * D_DIM);

    // 1) XW = x @ W  (WMMA f32 16x16x4)
    const int rowTiles = (N + 15) / 16;
    dim3 ggrid((rowTiles + 7) / 8, D_DIM / 16);
    gemm_xw_wmma<<<ggrid, 256, 0, stream>>>(x, weight, xw, N);

    // 2) symmetric weighted scatter (memory-bound core of the layer)
    edge_scatter<<<4096, 256, 0, stream>>>(xw, edge_index, edge_type, alpha,
                                           out, E, N, SELF_REL);

    // 3) column batch statistics
    col_stats<<<512, 256, 0, stream>>>(out, sums, sumsq, N);

    // 4) normalize in place
    bn_normalize<<<2048, 256, 0, stream>>>(out, sums, sumsq, gamma, beta, N);

    // 5) r passthrough -> second tuple element, concatenated after out
    hipMemcpyAsync(out + nOut, r, (size_t)in_sizes[1] * sizeof(float),
                   hipMemcpyDeviceToDevice, stream);
}